// StructuralFeatureLayer_34600256537309
// MI455X (gfx1250) — compile-verified
//
#include <hip/hip_runtime.h>
#include <hip/hip_bf16.h>

typedef __attribute__((ext_vector_type(2))) float v2f;
typedef __attribute__((ext_vector_type(8))) float v8f;

#define N_NODES 100000
#define CH      128      // in/out channels
#define MTILES  (N_NODES / 16)   // 6250, exact

// ---------------------------------------------------------------------------
// Kernel 1: fused dual GEMM via V_WMMA_F32_16X16X4_F32 (wave32).
//   y_self = x @ W_S^T + b_S  -> out_self (d_out)
//   y_nbr  = x @ W_N^T + b_N  -> out_nbr  (d_ws)
// Each wave: one 16-row node tile + one weight matrix; accumulates all
// eight 16-wide N-tiles (8 x v8f accumulators) over K=128 in steps of 4.
//
// Fragment layouts (CDNA5 ISA 7.12.2, fp32):
//   A 16x4 : lane(0-15)=M, lane(16-31)=M; VGPR0={K0|K2}, VGPR1={K1|K3}
//            -> lane loads float2 at x[row*128 + k0 + 2*half]
//   B 4x16 : row-striped across lanes; B[k][n] = W[n][k] (W row-major OxI)
//            -> lane loads float2 at W[(n0+lh)*128 + k0 + 2*half]
//   C 16x16: VGPR r holds M = r + 8*half, N = lane&15
// ---------------------------------------------------------------------------
__global__ void __launch_bounds__(256)
sfl_gemm_wmma(const float* __restrict__ x,
              const float* __restrict__ Ws, const float* __restrict__ bS,
              const float* __restrict__ Wn, const float* __restrict__ bN,
              float* __restrict__ out_self, float* __restrict__ out_nbr)
{
    const int wave = (int)((blockIdx.x * blockDim.x + threadIdx.x) >> 5);
    if (wave >= MTILES * 2) return;            // wave-uniform: EXEC stays all-1s
    const int lane  = (int)(threadIdx.x & 31);
    const int mat   = wave & 1;                // 0 -> W_S, 1 -> W_N
    const int mtile = wave >> 1;

    const float* __restrict__ W    = mat ? Wn : Ws;
    const float* __restrict__ bias = mat ? bN : bS;
    float*       __restrict__ outp = mat ? out_nbr : out_self;

    const int half = lane >> 4;                // 0: K{0,1} / M 0-7 ; 1: K{2,3} / M 8-15
    const int lh   = lane & 15;

    v8f acc[8];
    const v8f zero = {0.f,0.f,0.f,0.f,0.f,0.f,0.f,0.f};
#pragma unroll
    for (int t = 0; t < 8; ++t) acc[t] = zero;

    const float* __restrict__ arow = x + (size_t)(mtile * 16 + lh) * CH;

    for (int k0 = 0; k0 < CH; k0 += 4) {
        const v2f a = *(const v2f*)(arow + k0 + 2 * half);      // A fragment
#pragma unroll
        for (int t = 0; t < 8; ++t) {
            const int n0 = t * 16;
            const v2f b = *(const v2f*)(W + (size_t)(n0 + lh) * CH + k0 + 2 * half);
            acc[t] = __builtin_amdgcn_wmma_f32_16x16x4_f32(
                false, a, false, b, (short)0, acc[t], false, false);
        }
    }

#pragma unroll
    for (int t = 0; t < 8; ++t) {
        const int n0 = t * 16;
        const float bv = bias[n0 + lh];
#pragma unroll
        for (int r = 0; r < 8; ++r) {
            const int m = mtile * 16 + r + 8 * half;
            outp[(size_t)m * CH + n0 + lh] = acc[t][r] + bv;
        }
    }
}

// ---------------------------------------------------------------------------
// Kernel 2: edge scatter. One wave per edge; lane handles 4 channels.
//   d_out[row] += exp(-td) * y_nbr[col]   (204M global_atomic_add_f32 total)
// Gather of y_nbr[col] is a coalesced 512B burst per edge.
// ---------------------------------------------------------------------------
__global__ void __launch_bounds__(256)
sfl_edge_scatter(const int* __restrict__ rows, const int* __restrict__ cols,
                 const float* __restrict__ td, const float* __restrict__ ynbr,
                 float* __restrict__ out, int E)
{
    const long long gid = (long long)blockIdx.x * blockDim.x + threadIdx.x;
    const int e = (int)(gid >> 5);
    if (e >= E) return;
    const int lane = (int)(gid & 31);

    const int r = rows[e];
    const int c = cols[e];
    const float w = __expf(-td[e]);            // TAU = 1.0

    const float4 v = *(const float4*)(ynbr + (size_t)c * CH + lane * 4);
    float* dst = out + (size_t)r * CH + lane * 4;
    atomicAdd(dst + 0, v.x * w);
    atomicAdd(dst + 1, v.y * w);
    atomicAdd(dst + 2, v.z * w);
    atomicAdd(dst + 3, v.w * w);
}

// ---------------------------------------------------------------------------
// Kernel 3: in-place ReLU over d_out (float4 vectorized).
// ---------------------------------------------------------------------------
__global__ void __launch_bounds__(256)
sfl_relu(float4* __restrict__ out, int n4)
{
    const int i = (int)(blockIdx.x * blockDim.x + threadIdx.x);
    if (i >= n4) return;
    float4 v = out[i];
    v.x = fmaxf(v.x, 0.f);
    v.y = fmaxf(v.y, 0.f);
    v.z = fmaxf(v.z, 0.f);
    v.w = fmaxf(v.w, 0.f);
    out[i] = v;
}

extern "C" void kernel_launch(void* const* d_in, const int* in_sizes, int n_in,
                              void* d_out, int out_size, void* d_ws, size_t ws_size,
                              hipStream_t stream)
{
    const float* x    = (const float*)d_in[0];
    const int*   eidx = (const int*)  d_in[1];   // (2, E) flat: [rows | cols]
    const float* td   = (const float*)d_in[2];
    const float* WSw  = (const float*)d_in[3];
    const float* WSb  = (const float*)d_in[4];
    const float* WNw  = (const float*)d_in[5];
    const float* WNb  = (const float*)d_in[6];

    const int E = in_sizes[2];                   // 1,600,000
    float* out  = (float*)d_out;                 // N_NODES * 128
    float* ynbr = (float*)d_ws;                  // N_NODES * 128 floats (51.2 MB)

    // 1) Dual GEMM: self features -> d_out, neighbor features -> d_ws.
    {
        const int waves  = MTILES * 2;           // 12,500
        const int blocks = (waves * 32 + 255) / 256;
        sfl_gemm_wmma<<<blocks, 256, 0, stream>>>(x, WSw, WSb, WNw, WNb, out, ynbr);
    }

    // 2) Edge gather/scale/scatter-add into d_out.
    {
        const long long threads = (long long)E * 32;
        const int blocks = (int)((threads + 255) / 256);
        sfl_edge_scatter<<<blocks, 256, 0, stream>>>(eidx, eidx + E, td, ynbr, out, E);
    }

    // 3) ReLU in place.
    {
        const int n4 = (N_NODES * CH) / 4;       // 3,200,000
        const int blocks = (n4 + 255) / 256;
        sfl_relu<<<blocks, 256, 0, stream>>>((float4*)out, n4);
    }
}